// Mask_CID_41360535060715
// MI455X (gfx1250) — compile-verified
//
#include <hip/hip_runtime.h>

typedef __attribute__((ext_vector_type(2))) float v2f;
typedef __attribute__((ext_vector_type(8))) float v8f;

// One wave32 per sample. For each 16-capsule tile:
//   A (16x4 f32) = squared elements of 16 capsule rows (K-chunked by 4),
//   B (4x16 f32) = all-ones  =>  D[m,n] = sum_k A[m,k]  (row sum of squares).
// Accumulate 16 WMMA steps to cover D=64, giving norms^2 for 16 capsules.
// D layout: acc[v] on lanes 0-15 = ||x[cb+v]||^2, lanes 16-31 = ||x[cb+v+8]||^2.
__global__ __launch_bounds__(256, 1)
void caps_norm_argmax_mask(const float* __restrict__ x,
                           float* __restrict__ out_masked,
                           float* __restrict__ out_idx,
                           int nBatch)
{
    constexpr int C  = 100;
    constexpr int D  = 64;
    constexpr int CS = C * D;          // 6400 floats per sample

    const int lane = threadIdx.x & 31;
    const int wave = threadIdx.x >> 5;
    const int b    = blockIdx.x * 8 + wave;   // wave-uniform
    if (b >= nBatch) return;                  // whole-wave exit, EXEC stays full

    const float* xb  = x + (size_t)b * CS;
    const int half   = lane >> 4;             // 0: K={0,1}, 1: K={2,3} per WMMA step
    const int mrow   = lane & 15;             // A-matrix row M

    const v2f onesB  = {1.0f, 1.0f};          // 4x16 all-ones B operand
    float bestVal = -1.0f;                    // norms^2 >= 0, so -1 never wins
    int   bestIdx = 0;

    for (int cb = 0; cb < C; cb += 16) {
        const int   crow  = cb + mrow;
        const int   csafe = crow < C ? crow : (C - 1);   // clamp: never read OOB
        const float valid = crow < C ? 1.0f : 0.0f;      // zero padded rows
        const float* rowp = xb + csafe * D + 2 * half;

        v8f acc = {};
#pragma unroll
        for (int k = 0; k < 16; ++k) {
            v2f v = *(const v2f*)(rowp + 4 * k);         // 8B coalesced load
            v2f a;
            a.x = v.x * v.x * valid;
            a.y = v.y * v.y * valid;
            // v_wmma_f32_16x16x4_f32 : D = A x B + C
            acc = __builtin_amdgcn_wmma_f32_16x16x4_f32(
                false, a, false, onesB, (short)0, acc, false, false);
        }

        // local argmax over this tile's 8 norms held by this lane (branchless)
#pragma unroll
        for (int v = 0; v < 8; ++v) {
            const int   idx = cb + v + 8 * half;
            const float val = acc[v];
            bool take = (idx < C) &&
                        ((val > bestVal) || (val == bestVal && idx < bestIdx));
            bestVal = take ? val : bestVal;
            bestIdx = take ? idx : bestIdx;
        }
    }

    // wave32 butterfly argmax reduction (tie-break: lowest index, like jnp.argmax)
#pragma unroll
    for (int m = 16; m >= 1; m >>= 1) {
        float ov = __shfl_xor(bestVal, m, 32);
        int   oi = __shfl_xor(bestIdx, m, 32);
        bool take = (ov > bestVal) || (ov == bestVal && oi < bestIdx);
        bestVal = take ? ov : bestVal;
        bestIdx = take ? oi : bestIdx;
    }

    // gather winning capsule (64 floats: 2 per lane, 8B stores) — L2-hot re-read
    const float* src = xb + bestIdx * D;
    v2f out2 = *(const v2f*)(src + lane * 2);
    *(v2f*)(out_masked + (size_t)b * D + lane * 2) = out2;
    if (lane == 0) out_idx[b] = (float)bestIdx;
}

extern "C" void kernel_launch(void* const* d_in, const int* in_sizes, int n_in,
                              void* d_out, int out_size, void* d_ws, size_t ws_size,
                              hipStream_t stream) {
    (void)n_in; (void)out_size; (void)d_ws; (void)ws_size;
    const float* x = (const float*)d_in[0];
    const int nBatch = in_sizes[0] / (100 * 64);          // 16384

    float* out_masked = (float*)d_out;                    // [B, 1, 64] flat
    float* out_idx    = (float*)d_out + (size_t)nBatch * 64;  // [B] indices

    const int wavesPerBlock = 8;                          // 256 threads = 8 wave32
    dim3 grid((nBatch + wavesPerBlock - 1) / wavesPerBlock);
    caps_norm_argmax_mask<<<grid, 256, 0, stream>>>(x, out_masked, out_idx, nBatch);
}